// HeteroGraphSAGE_PEARL_24524263260183
// MI455X (gfx1250) — compile-verified
//
#include <hip/hip_runtime.h>
#include <hip/hip_bf16.h>

#define C_DIM 128
#define PANEL_ROWS 64
#define LDS_STRIDE (C_DIM + 4)   // 132 floats: 16B-aligned rows, banks 4 apart

typedef float v2f __attribute__((ext_vector_type(2)));
typedef float v8f __attribute__((ext_vector_type(8)));

// ---------------------------------------------------------------------------
// Zero workspace (agg sums + degree counters), grid-stride.
// ---------------------------------------------------------------------------
__global__ void hgs_zero(float* __restrict__ p, size_t n) {
    size_t i = (size_t)blockIdx.x * blockDim.x + threadIdx.x;
    size_t stride = (size_t)gridDim.x * blockDim.x;
    for (; i < n; i += stride) p[i] = 0.0f;
}

// ---------------------------------------------------------------------------
// Edge scatter: one wave (32 lanes) per edge, 4 channels per lane (float4
// gather), native f32 global atomics for the per-destination sum; lane 0
// accumulates the in-degree.
// ---------------------------------------------------------------------------
__global__ void hgs_scatter_add(const float* __restrict__ xsrc,
                                const int* __restrict__ src,
                                const int* __restrict__ dst,
                                float* __restrict__ agg,
                                float* __restrict__ cnt,
                                int nE) {
    const int lane = threadIdx.x & 31;
    const int e = blockIdx.x * 8 + (threadIdx.x >> 5);
    if (e >= nE) return;
    const int s = src[e];
    const int d = dst[e];
    const float4 v = ((const float4*)(xsrc + (size_t)s * C_DIM))[lane];
    float* base = agg + (size_t)d * C_DIM + lane * 4;
    unsafeAtomicAdd(base + 0, v.x);
    unsafeAtomicAdd(base + 1, v.y);
    unsafeAtomicAdd(base + 2, v.z);
    unsafeAtomicAdd(base + 3, v.w);
    if (lane == 0) unsafeAtomicAdd(cnt + d, 1.0f);
}

// ---------------------------------------------------------------------------
// Fused SAGE linear: new = (agg/max(cnt,1)) @ Wl + bl + x @ Wr
//
// Block: 256 threads = 8 waves, covers 64 rows x 128 cols of output.
// Wave w owns columns [16w, 16w+16) and iterates 4 M-tiles (4 accumulators),
// so each B-operand fetch feeds 4 v_wmma_f32_16x16x4_f32 ops.
// A panels staged through padded LDS (conflict-free across M rows) with the
// seg-mean 1/max(deg,1) scale applied at staging time; B read from global
// (the two 64KB weight matrices are L2-resident across all 1563 blocks).
//
// A layout (16x4 f32): lanes 0-15 -> M=lane, VGPRs = K {0,1};
//                      lanes 16-31 -> M=lane-16, VGPRs = K {2,3}.
// B layout (4x16 f32): lanes 0-15 -> N=lane, rows K {0,1};
//                      lanes 16-31 -> N=lane-16, rows K {2,3}.
// C/D (16x16 f32): VGPR i, lanes 0-15 -> M=i; lanes 16-31 -> M=i+8.
// ---------------------------------------------------------------------------
__global__ void hgs_gemm_tile(const float* __restrict__ agg,
                              const float* __restrict__ cnt,
                              const float* __restrict__ xin,
                              const float* __restrict__ Wl,
                              const float* __restrict__ bias,
                              const float* __restrict__ Wr,
                              float* __restrict__ outp,
                              int n) {
    __shared__ float s_a[PANEL_ROWS * LDS_STRIDE];  // mean-scaled aggregates
    __shared__ float s_x[PANEL_ROWS * LDS_STRIDE];  // destination features

    const int tid = threadIdx.x;
    const int row0 = blockIdx.x * PANEL_ROWS;

    // Cooperative staging: 64 rows x 128 cols, float4 granularity.
    for (int i = tid; i < PANEL_ROWS * (C_DIM / 4); i += 256) {
        const int r = i >> 5;          // row within panel (C_DIM/4 == 32)
        const int c4 = i & 31;         // float4 column
        int row = row0 + r;
        if (row >= n) row = n - 1;     // clamp tail rows (stores are guarded)
        const float scale = 1.0f / fmaxf(cnt[row], 1.0f);
        const float4 va = ((const float4*)(agg + (size_t)row * C_DIM))[c4];
        const float4 vx = ((const float4*)(xin + (size_t)row * C_DIM))[c4];
        float4 sa;
        sa.x = va.x * scale; sa.y = va.y * scale;
        sa.z = va.z * scale; sa.w = va.w * scale;
        ((float4*)(s_a + r * LDS_STRIDE))[c4] = sa;
        ((float4*)(s_x + r * LDS_STRIDE))[c4] = vx;
    }
    __syncthreads();

    const int wave = tid >> 5;
    const int lane = tid & 31;
    const int half = lane >> 4;        // 0: K+{0,1}, D rows 0-7; 1: K+{2,3}, rows 8-15
    const int l16 = lane & 15;
    const int colN = wave * 16 + l16;  // output column this lane holds in B/D

    const float* __restrict__ wl_col = Wl + colN;
    const float* __restrict__ wr_col = Wr + colN;

    v8f c0 = {}, c1 = {}, c2 = {}, c3 = {};

    // lin_l: mean-aggregated neighbors @ Wl (B reused across 4 M-tiles)
#pragma unroll 4
    for (int k = 0; k < C_DIM; k += 4) {
        const int kk = k + half * 2;
        v2f b;
        b.x = wl_col[(size_t)kk * C_DIM];
        b.y = wl_col[(size_t)(kk + 1) * C_DIM];
        const float* sa = s_a + l16 * LDS_STRIDE + kk;
        v2f a;
        a.x = sa[0];               a.y = sa[1];
        c0 = __builtin_amdgcn_wmma_f32_16x16x4_f32(false, a, false, b, (short)0, c0, false, false);
        a.x = sa[16 * LDS_STRIDE]; a.y = sa[16 * LDS_STRIDE + 1];
        c1 = __builtin_amdgcn_wmma_f32_16x16x4_f32(false, a, false, b, (short)0, c1, false, false);
        a.x = sa[32 * LDS_STRIDE]; a.y = sa[32 * LDS_STRIDE + 1];
        c2 = __builtin_amdgcn_wmma_f32_16x16x4_f32(false, a, false, b, (short)0, c2, false, false);
        a.x = sa[48 * LDS_STRIDE]; a.y = sa[48 * LDS_STRIDE + 1];
        c3 = __builtin_amdgcn_wmma_f32_16x16x4_f32(false, a, false, b, (short)0, c3, false, false);
    }
    // lin_r: destination features @ Wr
#pragma unroll 4
    for (int k = 0; k < C_DIM; k += 4) {
        const int kk = k + half * 2;
        v2f b;
        b.x = wr_col[(size_t)kk * C_DIM];
        b.y = wr_col[(size_t)(kk + 1) * C_DIM];
        const float* sx = s_x + l16 * LDS_STRIDE + kk;
        v2f a;
        a.x = sx[0];               a.y = sx[1];
        c0 = __builtin_amdgcn_wmma_f32_16x16x4_f32(false, a, false, b, (short)0, c0, false, false);
        a.x = sx[16 * LDS_STRIDE]; a.y = sx[16 * LDS_STRIDE + 1];
        c1 = __builtin_amdgcn_wmma_f32_16x16x4_f32(false, a, false, b, (short)0, c1, false, false);
        a.x = sx[32 * LDS_STRIDE]; a.y = sx[32 * LDS_STRIDE + 1];
        c2 = __builtin_amdgcn_wmma_f32_16x16x4_f32(false, a, false, b, (short)0, c2, false, false);
        a.x = sx[48 * LDS_STRIDE]; a.y = sx[48 * LDS_STRIDE + 1];
        c3 = __builtin_amdgcn_wmma_f32_16x16x4_f32(false, a, false, b, (short)0, c3, false, false);
    }

    const float bv = bias[colN];
#pragma unroll
    for (int i = 0; i < 8; ++i) {
        const int rbase = row0 + half * 8 + i;
        int row;
        row = rbase;      if (row < n) outp[(size_t)row * C_DIM + colN] = c0[i] + bv;
        row = rbase + 16; if (row < n) outp[(size_t)row * C_DIM + colN] = c1[i] + bv;
        row = rbase + 32; if (row < n) outp[(size_t)row * C_DIM + colN] = c2[i] + bv;
        row = rbase + 48; if (row < n) outp[(size_t)row * C_DIM + colN] = c3[i] + bv;
    }
}

// ---------------------------------------------------------------------------
// LayerNorm + ReLU: one wave per node row (C=128 -> 4 channels/lane),
// shfl-xor tree reduction for mean / E[x^2].
// ---------------------------------------------------------------------------
__global__ void hgs_ln_relu(const float* __restrict__ inp,
                            const float* __restrict__ w,
                            const float* __restrict__ b,
                            float* __restrict__ outp,
                            int n) {
    const int lane = threadIdx.x & 31;
    const int row = blockIdx.x * 8 + (threadIdx.x >> 5);
    if (row >= n) return;
    const float4 v = ((const float4*)(inp + (size_t)row * C_DIM))[lane];
    float s = v.x + v.y + v.z + v.w;
    float s2 = v.x * v.x + v.y * v.y + v.z * v.z + v.w * v.w;
#pragma unroll
    for (int m = 16; m >= 1; m >>= 1) {
        s += __shfl_xor(s, m, 32);
        s2 += __shfl_xor(s2, m, 32);
    }
    const float mu = s * (1.0f / C_DIM);
    const float var = s2 * (1.0f / C_DIM) - mu * mu;
    const float inv = rsqrtf(var + 1e-5f);
    const int c0 = lane * 4;
    float4 o;
    o.x = fmaxf(0.0f, (v.x - mu) * inv * w[c0 + 0] + b[c0 + 0]);
    o.y = fmaxf(0.0f, (v.y - mu) * inv * w[c0 + 1] + b[c0 + 1]);
    o.z = fmaxf(0.0f, (v.z - mu) * inv * w[c0 + 2] + b[c0 + 2]);
    o.w = fmaxf(0.0f, (v.w - mu) * inv * w[c0 + 3] + b[c0 + 3]);
    ((float4*)(outp + (size_t)row * C_DIM))[lane] = o;
}

// ---------------------------------------------------------------------------
// Host orchestration (graph-capture safe: kernels on `stream` only).
// ---------------------------------------------------------------------------
extern "C" void kernel_launch(void* const* d_in, const int* in_sizes, int n_in,
                              void* d_out, int out_size, void* d_ws, size_t ws_size,
                              hipStream_t stream) {
    const float* x_user = (const float*)d_in[0];
    const float* x_item = (const float*)d_in[1];
    // d_in[2] = PE (unused by the reference)
    const float* Wl = (const float*)d_in[3];     // [L,2,C,C]
    const float* bl = (const float*)d_in[4];     // [L,2,C]
    const float* Wr = (const float*)d_in[5];     // [L,2,C,C]
    const float* ln_w = (const float*)d_in[6];   // [L,2,C]
    const float* ln_b = (const float*)d_in[7];   // [L,2,C]
    const int* e_ui_src = (const int*)d_in[8];
    const int* e_ui_dst = (const int*)d_in[9];
    const int* e_iu_src = (const int*)d_in[10];
    const int* e_iu_dst = (const int*)d_in[11];
    // d_in[12] = reverse_node_mapping (unused by the reference)

    const int N = in_sizes[0] / C_DIM;   // 100000 (users == items here)
    const int E = in_sizes[8];           // 640000

    float* out = (float*)d_out;
    float* xu_out = out;                     // [N,C]
    float* xi_out = out + (size_t)N * C_DIM; // [N,C]

    float* ws = (float*)d_ws;
    float* agg_u = ws;                               // [N,C]
    float* agg_i = agg_u + (size_t)N * C_DIM;        // [N,C]
    float* cnt_u = agg_i + (size_t)N * C_DIM;        // [N]
    float* cnt_i = cnt_u + N;                        // [N]
    float* new_u = cnt_i + N;                        // [N,C]
    float* new_i = new_u + (size_t)N * C_DIM;        // [N,C]

    const size_t zero_count = 2 * (size_t)N * C_DIM + 2 * (size_t)N;
    const int scatter_blocks = (E + 7) / 8;
    const int gemm_blocks = (N + PANEL_ROWS - 1) / PANEL_ROWS;
    const int ln_blocks = (N + 7) / 8;

    const float* xu_cur = x_user;
    const float* xi_cur = x_item;

    for (int l = 0; l < 2; ++l) {
        hgs_zero<<<2048, 256, 0, stream>>>(agg_u, zero_count);

        // user -> item aggregation, item -> user aggregation
        hgs_scatter_add<<<scatter_blocks, 256, 0, stream>>>(
            xu_cur, e_ui_src, e_ui_dst, agg_i, cnt_i, E);
        hgs_scatter_add<<<scatter_blocks, 256, 0, stream>>>(
            xi_cur, e_iu_src, e_iu_dst, agg_u, cnt_u, E);

        // new_i = mean(agg_i) @ Wl[l,0] + bl[l,0] + x_item @ Wr[l,0]
        hgs_gemm_tile<<<gemm_blocks, 256, 0, stream>>>(
            agg_i, cnt_i, xi_cur,
            Wl + ((size_t)l * 2 + 0) * C_DIM * C_DIM,
            bl + ((size_t)l * 2 + 0) * C_DIM,
            Wr + ((size_t)l * 2 + 0) * C_DIM * C_DIM,
            new_i, N);
        // new_u = mean(agg_u) @ Wl[l,1] + bl[l,1] + x_user @ Wr[l,1]
        hgs_gemm_tile<<<gemm_blocks, 256, 0, stream>>>(
            agg_u, cnt_u, xu_cur,
            Wl + ((size_t)l * 2 + 1) * C_DIM * C_DIM,
            bl + ((size_t)l * 2 + 1) * C_DIM,
            Wr + ((size_t)l * 2 + 1) * C_DIM * C_DIM,
            new_u, N);

        // x_user = relu(LN(new_u; ln[l,0])), x_item = relu(LN(new_i; ln[l,1]))
        hgs_ln_relu<<<ln_blocks, 256, 0, stream>>>(
            new_u, ln_w + ((size_t)l * 2 + 0) * C_DIM,
            ln_b + ((size_t)l * 2 + 0) * C_DIM, xu_out, N);
        hgs_ln_relu<<<ln_blocks, 256, 0, stream>>>(
            new_i, ln_w + ((size_t)l * 2 + 1) * C_DIM,
            ln_b + ((size_t)l * 2 + 1) * C_DIM, xi_out, N);

        xu_cur = xu_out;
        xi_cur = xi_out;
    }
}